// ResHybNet_19353122636487
// MI455X (gfx1250) — compile-verified
//
#include <hip/hip_runtime.h>
#include <hip/hip_bf16.h>

typedef __attribute__((ext_vector_type(2))) float v2f;
typedef __attribute__((ext_vector_type(8))) float v8f;

#define DIMS 10
#define HID 16

// ---------------- degree / dinv ----------------
__global__ void k_init(int* deg, float* g1, float* g2, int n) {
    int i = blockIdx.x * blockDim.x + threadIdx.x;
    if (i < n) deg[i] = 1;                 // self loop
    for (int t = i; t < n * HID; t += gridDim.x * blockDim.x) g1[t] = 0.f;
    for (int t = i; t < n * DIMS; t += gridDim.x * blockDim.x) g2[t] = 0.f;
}

__global__ void k_count_deg(const int* dst, int* deg, int E) {
    int e = blockIdx.x * blockDim.x + threadIdx.x;
    if (e < E) atomicAdd(&deg[dst[e]], 1);
}

__global__ void k_dinv(const int* deg, float* dinv, int n) {
    int i = blockIdx.x * blockDim.x + threadIdx.x;
    if (i < n) dinv[i] = rsqrtf((float)deg[i]);
}

// ---------------- WMMA tiled GEMM: h1 = x @ W1  (Nx10 @ 10x16) ----------------
// V_WMMA_F32_16X16X4_F32 layout:
//   A: lanes 0-15 -> M=lane, {K0,K1}; lanes 16-31 -> M=lane-16, {K2,K3}
//   B: lanes 0-15 -> N=lane, {K0,K1}; lanes 16-31 -> N=lane-16, {K2,K3}
//   D: VGPR j: lanes0-15 (M=j,N=lane), lanes16-31 (M=j+8,N=lane-16)
// All loads unconditional (clamped addresses); invalid data zeroed via selects.
__global__ void k_gemm1(const float* __restrict__ x, const float* __restrict__ W1,
                        float* __restrict__ h1, int n) {
    int wave = (blockIdx.x * blockDim.x + threadIdx.x) >> 5;
    int lane = threadIdx.x & 31;
    int row0 = wave * 16;
    if (row0 >= n) return;                 // wave-uniform exit
    int m  = lane & 15;
    int hi = lane >> 4;
    int row = row0 + m;
    bool rv = row < n;
    int rowc = rv ? row : (n - 1);

    const float* xr = x + (size_t)rowc * DIMS;
    v2f a0 = *(const v2f*)(xr + 2 * hi);       // K 0..3
    v2f a1 = *(const v2f*)(xr + 4 + 2 * hi);   // K 4..7
    v2f a2 = *(const v2f*)(xr + 8);            // K 8..9 (always in-bounds)
    v2f zz = {0.f, 0.f};
    if (!rv) { a0 = zz; a1 = zz; }
    if (!rv || hi) a2 = zz;                    // K 10..11 padded

    v2f b0, b1, b2;
    b0.x = W1[(0 + 2 * hi) * HID + m];  b0.y = W1[(1 + 2 * hi) * HID + m];
    b1.x = W1[(4 + 2 * hi) * HID + m];  b1.y = W1[(5 + 2 * hi) * HID + m];
    b2.x = W1[8 * HID + m];             b2.y = W1[9 * HID + m];
    if (hi) b2 = zz;

    v8f c = {};
    c = __builtin_amdgcn_wmma_f32_16x16x4_f32(false, a0, false, b0, (short)0, c, false, false);
    c = __builtin_amdgcn_wmma_f32_16x16x4_f32(false, a1, false, b1, (short)0, c, false, false);
    c = __builtin_amdgcn_wmma_f32_16x16x4_f32(false, a2, false, b2, (short)0, c, false, false);

    if (row0 + 16 <= n) {
        #pragma unroll
        for (int j = 0; j < 8; ++j) h1[(row0 + j + 8 * hi) * HID + m] = c[j];
    } else {
        #pragma unroll
        for (int j = 0; j < 8; ++j) {
            int r = row0 + j + 8 * hi;
            if (r < n) h1[r * HID + m] = c[j];
        }
    }
}

// h2 = g1 @ W2  (Nx16 @ 16x10, output padded to 16 cols)
__global__ void k_gemm2(const float* __restrict__ g1, const float* __restrict__ W2,
                        float* __restrict__ h2, int n) {
    int wave = (blockIdx.x * blockDim.x + threadIdx.x) >> 5;
    int lane = threadIdx.x & 31;
    int row0 = wave * 16;
    if (row0 >= n) return;
    int m  = lane & 15;
    int hi = lane >> 4;
    int row = row0 + m;
    bool rv = row < n;
    int rowc = rv ? row : (n - 1);
    int mc   = (m < DIMS) ? m : (DIMS - 1);

    const float* gr = g1 + (size_t)rowc * HID;
    v2f zz = {0.f, 0.f};
    v2f a[4], b[4];
    #pragma unroll
    for (int s = 0; s < 4; ++s) {
        a[s] = *(const v2f*)(gr + 4 * s + 2 * hi);
        int k = 4 * s + 2 * hi;
        b[s].x = W2[k * DIMS + mc];
        b[s].y = W2[(k + 1) * DIMS + mc];
    }
    #pragma unroll
    for (int s = 0; s < 4; ++s) {
        if (!rv)      a[s] = zz;
        if (m >= DIMS) b[s] = zz;
    }

    v8f c = {};
    #pragma unroll
    for (int s = 0; s < 4; ++s)
        c = __builtin_amdgcn_wmma_f32_16x16x4_f32(false, a[s], false, b[s], (short)0, c, false, false);

    if (row0 + 16 <= n) {
        #pragma unroll
        for (int j = 0; j < 8; ++j) h2[(row0 + j + 8 * hi) * HID + m] = c[j];
    } else {
        #pragma unroll
        for (int j = 0; j < 8; ++j) {
            int r = row0 + j + 8 * hi;
            if (r < n) h2[r * HID + m] = c[j];
        }
    }
}

// ---------------- edge scatter: out[d] += dinv[s]*dinv[d]*h[s] ----------------
template <int F, int HS, int OS>
__global__ void k_scatter(const int* __restrict__ src, const int* __restrict__ dst,
                          const float* __restrict__ dinv, const float* __restrict__ h,
                          float* __restrict__ out, int E) {
    int e = blockIdx.x * blockDim.x + threadIdx.x;
    if (e >= E) return;
    int s = src[e], d = dst[e];
    float w = dinv[s] * dinv[d];
    const float* hp = h + (size_t)s * HS;
    float* op = out + (size_t)d * OS;
    float msg[F];
    #pragma unroll
    for (int f = 0; f < F; ++f) msg[f] = w * hp[f];
    #pragma unroll
    for (int f = 0; f < F; ++f) unsafeAtomicAdd(op + f, msg[f]);
}

// g1 = relu(g1 + dinv^2*h1 + b1)
__global__ void k_fin1(float* g1, const float* h1, const float* dinv,
                       const float* b1, int n) {
    int t = blockIdx.x * blockDim.x + threadIdx.x;
    if (t >= n * HID) return;
    int i = t >> 4, f = t & 15;
    float di = dinv[i];
    float v = g1[t] + di * di * h1[t] + b1[f];
    g1[t] = v > 0.f ? v : 0.f;
}

// xdual (in g2) = x + g2 + dinv^2*h2 + b2
__global__ void k_fin2(float* g2, const float* h2, const float* x,
                       const float* dinv, const float* b2, int n) {
    int t = blockIdx.x * blockDim.x + threadIdx.x;
    if (t >= n * DIMS) return;
    int i = t / DIMS, f = t - i * DIMS;
    float di = dinv[i];
    g2[t] = x[t] + g2[t] + di * di * h2[i * HID + f] + b2[f];
}

// ---------------- fused conv stack + head + softmax ----------------
#define S_CW1 0
#define S_CB1 48
#define S_CW2 64
#define S_CB2 1600
#define S_WOUT 1632
#define S_BOUT 1760
#define S_TOT 1762

__global__ void k_conv(const float* __restrict__ xdual,
                       const float* __restrict__ cW1, const float* __restrict__ cb1,
                       const float* __restrict__ cW2, const float* __restrict__ cb2,
                       const float* __restrict__ Wout, const float* __restrict__ bout,
                       float* __restrict__ out, int n) {
    __shared__ float sm[S_TOT];
    for (int t = threadIdx.x; t < 48;   t += blockDim.x) sm[S_CW1 + t] = cW1[t];
    for (int t = threadIdx.x; t < 16;   t += blockDim.x) sm[S_CB1 + t] = cb1[t];
    for (int t = threadIdx.x; t < 1536; t += blockDim.x) sm[S_CW2 + t] = cW2[t];
    for (int t = threadIdx.x; t < 32;   t += blockDim.x) sm[S_CB2 + t] = cb2[t];
    for (int t = threadIdx.x; t < 128;  t += blockDim.x) sm[S_WOUT + t] = Wout[t];
    for (int t = threadIdx.x; t < 2;    t += blockDim.x) sm[S_BOUT + t] = bout[t];
    __syncthreads();

    int i = blockIdx.x * blockDim.x + threadIdx.x;
    if (i >= n) return;

    float xp[DIMS + 2];
    xp[0] = 0.f; xp[DIMS + 1] = 0.f;
    #pragma unroll
    for (int t = 0; t < DIMS; ++t) xp[t + 1] = xdual[i * DIMS + t];

    // conv1 (1->16, k=3, pad 1) + relu + maxpool2 -> p1[16][5]
    float p1[16][5];
    #pragma unroll
    for (int c = 0; c < 16; ++c) {
        float w0 = sm[S_CW1 + c * 3 + 0];
        float w1 = sm[S_CW1 + c * 3 + 1];
        float w2 = sm[S_CW1 + c * 3 + 2];
        float bb = sm[S_CB1 + c];
        #pragma unroll
        for (int p = 0; p < 5; ++p) {
            int t0 = 2 * p, t1 = 2 * p + 1;
            float ya = bb + w0 * xp[t0] + w1 * xp[t0 + 1] + w2 * xp[t0 + 2];
            float yb = bb + w0 * xp[t1] + w1 * xp[t1 + 1] + w2 * xp[t1 + 2];
            ya = ya > 0.f ? ya : 0.f;
            yb = yb > 0.f ? yb : 0.f;
            p1[c][p] = ya > yb ? ya : yb;
        }
    }

    // conv2 (16->32, k=3, pad 1) + relu + maxpool2, fused into head GEMM
    float o0 = sm[S_BOUT + 0], o1 = sm[S_BOUT + 1];
    for (int c2 = 0; c2 < 32; ++c2) {
        float z[5];
        #pragma unroll
        for (int p = 0; p < 5; ++p) {
            float acc = sm[S_CB2 + c2];
            #pragma unroll
            for (int ci = 0; ci < 16; ++ci) {
                const float* w = &sm[S_CW2 + (c2 * 16 + ci) * 3];
                float l = (p > 0) ? p1[ci][p - 1] : 0.f;
                float r = (p < 4) ? p1[ci][p + 1] : 0.f;
                acc += w[0] * l + w[1] * p1[ci][p] + w[2] * r;
            }
            z[p] = acc > 0.f ? acc : 0.f;
        }
        float m0 = z[0] > z[1] ? z[0] : z[1];
        float m1 = z[2] > z[3] ? z[2] : z[3];
        int r0 = c2 * 2, r1 = c2 * 2 + 1;
        o0 += m0 * sm[S_WOUT + r0 * 2 + 0] + m1 * sm[S_WOUT + r1 * 2 + 0];
        o1 += m0 * sm[S_WOUT + r0 * 2 + 1] + m1 * sm[S_WOUT + r1 * 2 + 1];
    }

    float mx = o0 > o1 ? o0 : o1;
    float e0 = __expf(o0 - mx), e1 = __expf(o1 - mx);
    float inv = 1.f / (e0 + e1);
    out[i * 2 + 0] = e0 * inv;
    out[i * 2 + 1] = e1 * inv;
}

// ---------------- host launcher ----------------
extern "C" void kernel_launch(void* const* d_in, const int* in_sizes, int n_in,
                              void* d_out, int out_size, void* d_ws, size_t ws_size,
                              hipStream_t stream) {
    const float* x    = (const float*)d_in[0];
    const int*   ei   = (const int*)d_in[1];
    const float* W1   = (const float*)d_in[2];
    const float* b1   = (const float*)d_in[3];
    const float* W2   = (const float*)d_in[4];
    const float* b2   = (const float*)d_in[5];
    const float* cW1  = (const float*)d_in[6];
    const float* cb1  = (const float*)d_in[7];
    const float* cW2  = (const float*)d_in[8];
    const float* cb2  = (const float*)d_in[9];
    const float* Wout = (const float*)d_in[10];
    const float* bout = (const float*)d_in[11];
    float* out = (float*)d_out;

    int n = in_sizes[0] / DIMS;
    int E = in_sizes[1] / 2;
    const int* src = ei;
    const int* dst = ei + E;

    // workspace layout (4B elements)
    char* ws = (char*)d_ws;
    int*   deg  = (int*)ws;                                        // n
    float* dinv = (float*)(ws + (size_t)n * 4);                    // n
    float* h1   = (float*)(ws + (size_t)n * 8);                    // n*16
    float* g1   = (float*)(ws + (size_t)n * 8 + (size_t)n * 64);   // n*16
    float* h2   = (float*)(ws + (size_t)n * 8 + (size_t)n * 128);  // n*16
    float* g2   = (float*)(ws + (size_t)n * 8 + (size_t)n * 192);  // n*10

    const int B = 256;
    int gbN  = (n + B - 1) / B;
    int gbE  = (E + B - 1) / B;
    int tiles = (n + 15) / 16;
    int gbT  = (tiles + (B / 32) - 1) / (B / 32);

    k_init<<<gbN, B, 0, stream>>>(deg, g1, g2, n);
    k_count_deg<<<gbE, B, 0, stream>>>(dst, deg, E);
    k_dinv<<<gbN, B, 0, stream>>>(deg, dinv, n);

    // layer 1
    k_gemm1<<<gbT, B, 0, stream>>>(x, W1, h1, n);
    k_scatter<HID, HID, HID><<<gbE, B, 0, stream>>>(src, dst, dinv, h1, g1, E);
    k_fin1<<<(n * HID + B - 1) / B, B, 0, stream>>>(g1, h1, dinv, b1, n);

    // layer 2
    k_gemm2<<<gbT, B, 0, stream>>>(g1, W2, h2, n);
    k_scatter<DIMS, HID, DIMS><<<gbE, B, 0, stream>>>(src, dst, dinv, h2, g2, E);
    k_fin2<<<(n * DIMS + B - 1) / B, B, 0, stream>>>(g2, h2, x, dinv, b2, n);

    // conv stack + head + softmax
    k_conv<<<gbN, B, 0, stream>>>(g2, cW1, cb1, cW2, cb2, Wout, bout, out, n);
}